// GaussianMultiViewMerge_5849745457885
// MI455X (gfx1250) — compile-verified
//
#include <hip/hip_runtime.h>
#include <math.h>

// ---------------------------------------------------------------------------
// GaussianMultiViewMerge for MI455X (gfx1250, wave32)
// Segment-reduce pipeline; memory-bound (~0.75 GB traffic -> ~32us @ 23.3TB/s)
// CDNA5 paths: ds_add_f32 LDS atomics, global_atomic_add_f32 (agent scope),
// GLOBAL_LOAD_ASYNC_TO_LDS_B128 + s_wait_asynccnt staging of mean2 slices.
// ---------------------------------------------------------------------------

namespace {
constexpr int BTN   = 8;         // B*T
constexpr int NPT   = 149760;    // V*H*W points per (b,t)
constexpr int KSEG  = 64;        // track slots per (b,t)
constexpr int SEGS  = BTN * KSEG;          // 512
constexpr int CH    = 50;
constexpr int CP1   = 51;                  // 50 channels + count
constexpr int LPAD  = 53;                  // odd LDS row stride (bank spread)
constexpr int BLK   = 256;
constexpr int TILES = 9;
constexpr int BLKX  = 65;                  // 65*9*256 == 149760
constexpr float THR2 = 4.0f;               // THR^2
constexpr float EPSN = 1e-12f;

// workspace layout (float offsets)
constexpr int WS_CNT1  = 0;                        // 512
constexpr int WS_SUM1  = SEGS;                     // 512*3
constexpr int WS_MEAN1 = SEGS * 4;                 // 512*3
constexpr int WS_SUM2  = SEGS * 7;                 // 512*51 (col 50 = count2)
constexpr int WS_MEAN2 = SEGS * 7 + SEGS * CP1;    // 512*51
constexpr int WS_ZERO  = WS_SUM2 + SEGS * CP1;     // floats to zero = 29696
}

#if __has_builtin(__builtin_amdgcn_global_load_async_to_lds_b128)
#define HAS_ASYNC_LDS 1
typedef int v4i __attribute__((ext_vector_type(4)));
typedef v4i __attribute__((address_space(1))) global_v4i;
typedef v4i __attribute__((address_space(3))) shared_v4i;
#endif

__device__ __forceinline__ void gatomicAdd(float* p, float v) {
    __hip_atomic_fetch_add(p, v, __ATOMIC_RELAXED, __HIP_MEMORY_SCOPE_AGENT);
}

__global__ __launch_bounds__(BLK) void k_zero(float* __restrict__ ws) {
    ws[blockIdx.x * BLK + threadIdx.x] = 0.0f;   // 116 * 256 == 29696
}

// Pass 1: per-segment count + center sum (LDS accumulate, one flush per block)
__global__ __launch_bounds__(BLK) void k_pass1(const float* __restrict__ center,
                                               const int* __restrict__ gid,
                                               float* __restrict__ cnt1,
                                               float* __restrict__ sum1) {
    __shared__ float lcnt[KSEG];
    __shared__ float lsum[KSEG * 3];
    const int tid = threadIdx.x;
    const int bt  = blockIdx.y;
    if (tid < KSEG) lcnt[tid] = 0.0f;
    if (tid < KSEG * 3) lsum[tid] = 0.0f;
    __syncthreads();

    const long base = (long)bt * NPT + (long)blockIdx.x * (TILES * BLK);
#pragma unroll
    for (int t = 0; t < TILES; ++t) {
        const long p  = base + t * BLK + tid;
        const int  id = gid[p];
        if (id >= 0 && id < KSEG) {
            const float* c = center + p * 3;
            atomicAdd(&lcnt[id], 1.0f);              // ds_add_f32
            atomicAdd(&lsum[id * 3 + 0], c[0]);
            atomicAdd(&lsum[id * 3 + 1], c[1]);
            atomicAdd(&lsum[id * 3 + 2], c[2]);
        }
    }
    __syncthreads();
    if (tid < KSEG)     gatomicAdd(&cnt1[bt * KSEG + tid], lcnt[tid]);
    if (tid < KSEG * 3) gatomicAdd(&sum1[bt * KSEG * 3 + tid], lsum[tid]);
}

__global__ __launch_bounds__(BLK) void k_mean1(const float* __restrict__ cnt1,
                                               const float* __restrict__ sum1,
                                               float* __restrict__ mean1) {
    const int s = blockIdx.x * BLK + threadIdx.x;
    if (s < SEGS) {
        const float den = fmaxf(cnt1[s], 1.0f);
        mean1[s * 3 + 0] = sum1[s * 3 + 0] / den;
        mean1[s * 3 + 1] = sum1[s * 3 + 1] / den;
        mean1[s * 3 + 2] = sum1[s * 3 + 2] / den;
    }
}

__device__ __forceinline__ bool point_sel(int id, int bt, long p,
                                          const float* __restrict__ center,
                                          const float* __restrict__ mean1,
                                          const float* __restrict__ cnt1,
                                          float c[3]) {
    if (id < 0 || id >= KSEG) return false;
    const int seg = bt * KSEG + id;
    c[0] = center[p * 3 + 0];
    c[1] = center[p * 3 + 1];
    c[2] = center[p * 3 + 2];
    const float n  = cnt1[seg];
    const float d0 = c[0] - mean1[seg * 3 + 0];
    const float d1 = c[1] - mean1[seg * 3 + 1];
    const float d2 = c[2] - mean1[seg * 3 + 2];
    const float dd = d0 * d0 + d1 * d1 + d2 * d2;
    return (n >= 2.0f) && (dd <= THR2);
}

// Pass 2: accumulate all 50 channels + count for selected points
__global__ __launch_bounds__(BLK) void k_pass2(const float* __restrict__ center,
                                               const float* __restrict__ offset,
                                               const float* __restrict__ opacity,
                                               const float* __restrict__ scale,
                                               const float* __restrict__ rotation,
                                               const float* __restrict__ feat_dc,
                                               const float* __restrict__ confidence,
                                               const float* __restrict__ aff,
                                               const float* __restrict__ motion,
                                               const int* __restrict__ gid,
                                               const float* __restrict__ mean1,
                                               const float* __restrict__ cnt1,
                                               float* __restrict__ sum2) {
    __shared__ float lacc[KSEG * LPAD];
    const int tid = threadIdx.x;
    const int bt  = blockIdx.y;
    for (int i = tid; i < KSEG * LPAD; i += BLK) lacc[i] = 0.0f;
    __syncthreads();

    const long base = (long)bt * NPT + (long)blockIdx.x * (TILES * BLK);
    for (int t = 0; t < TILES; ++t) {
        const long p  = base + t * BLK + tid;
        const int  id = gid[p];
        float c[3];
        if (point_sel(id, bt, p, center, mean1, cnt1, c)) {
            float* a = &lacc[id * LPAD];
            atomicAdd(a + 0, c[0]);
            atomicAdd(a + 1, c[1]);
            atomicAdd(a + 2, c[2]);
            atomicAdd(a + 3, offset[p * 3 + 0]);
            atomicAdd(a + 4, offset[p * 3 + 1]);
            atomicAdd(a + 5, offset[p * 3 + 2]);
            atomicAdd(a + 6, opacity[p]);
            atomicAdd(a + 7, scale[p * 3 + 0]);
            atomicAdd(a + 8, scale[p * 3 + 1]);
            atomicAdd(a + 9, scale[p * 3 + 2]);
            const float4 r = *reinterpret_cast<const float4*>(rotation + p * 4);
            atomicAdd(a + 10, r.x);
            atomicAdd(a + 11, r.y);
            atomicAdd(a + 12, r.z);
            atomicAdd(a + 13, r.w);
            atomicAdd(a + 14, feat_dc[p * 3 + 0]);
            atomicAdd(a + 15, feat_dc[p * 3 + 1]);
            atomicAdd(a + 16, feat_dc[p * 3 + 2]);
            atomicAdd(a + 17, confidence[p]);
            const float4* a4 = reinterpret_cast<const float4*>(aff + p * 16);
            const float4* m4 = reinterpret_cast<const float4*>(motion + p * 16);
#pragma unroll
            for (int q = 0; q < 4; ++q) {
                const float4 av = a4[q];
                atomicAdd(a + 18 + q * 4 + 0, av.x);
                atomicAdd(a + 18 + q * 4 + 1, av.y);
                atomicAdd(a + 18 + q * 4 + 2, av.z);
                atomicAdd(a + 18 + q * 4 + 3, av.w);
            }
#pragma unroll
            for (int q = 0; q < 4; ++q) {
                const float4 mv = m4[q];
                atomicAdd(a + 34 + q * 4 + 0, mv.x);
                atomicAdd(a + 34 + q * 4 + 1, mv.y);
                atomicAdd(a + 34 + q * 4 + 2, mv.z);
                atomicAdd(a + 34 + q * 4 + 3, mv.w);
            }
            atomicAdd(a + 50, 1.0f);  // count2
        }
    }
    __syncthreads();
    for (int i = tid; i < KSEG * CP1; i += BLK) {
        const int s = i / CP1, c = i % CP1;
        gatomicAdd(&sum2[(size_t)(bt * KSEG + s) * CP1 + c], lacc[s * LPAD + c]);
    }
}

// Finalize mean2 with quaternion renormalization (channels 10..13)
__global__ __launch_bounds__(64) void k_mean2(const float* __restrict__ sum2,
                                              float* __restrict__ mean2) {
    const int s = blockIdx.x;
    const int c = threadIdx.x;
    const float cnt = sum2[(size_t)s * CP1 + 50];
    const float den = fmaxf(cnt, 1.0f);
    if (c < CH) {
        float m = sum2[(size_t)s * CP1 + c] / den;
        if (c >= 10 && c < 14) {
            const float r0 = sum2[(size_t)s * CP1 + 10] / den;
            const float r1 = sum2[(size_t)s * CP1 + 11] / den;
            const float r2 = sum2[(size_t)s * CP1 + 12] / den;
            const float r3 = sum2[(size_t)s * CP1 + 13] / den;
            const float nr = sqrtf(r0 * r0 + r1 * r1 + r2 * r2 + r3 * r3);
            m = m / fmaxf(nr, EPSN);
        }
        mean2[(size_t)s * CP1 + c] = m;
    } else if (c == 50) {
        mean2[(size_t)s * CP1 + 50] = cnt;
    }
}

// Output: sel ? mean2[seg] : X   (mean2 bt-slice staged into LDS async)
__global__ __launch_bounds__(BLK) void k_out(const float* __restrict__ center,
                                             const float* __restrict__ offset,
                                             const float* __restrict__ opacity,
                                             const float* __restrict__ scale,
                                             const float* __restrict__ rotation,
                                             const float* __restrict__ feat_dc,
                                             const float* __restrict__ confidence,
                                             const float* __restrict__ aff,
                                             const float* __restrict__ motion,
                                             const int* __restrict__ gid,
                                             const float* __restrict__ mean1,
                                             const float* __restrict__ cnt1,
                                             const float* __restrict__ mean2,
                                             float* __restrict__ out) {
    __shared__ __align__(16) float lm[KSEG * CP1];   // 3264 floats = 816 x b128
    const int tid = threadIdx.x;
    const int bt  = blockIdx.y;
    const float* src = mean2 + (size_t)bt * KSEG * CP1;

#ifdef HAS_ASYNC_LDS
    for (int i = tid; i < (KSEG * CP1) / 4; i += BLK) {
        __builtin_amdgcn_global_load_async_to_lds_b128(
            (global_v4i*)(src + (size_t)i * 4),
            (shared_v4i*)(&lm[i * 4]), 0, 0);
    }
#if __has_builtin(__builtin_amdgcn_s_wait_asynccnt)
    __builtin_amdgcn_s_wait_asynccnt(0);
#else
    asm volatile("s_wait_asynccnt 0x0" ::: "memory");
#endif
#else
    for (int i = tid; i < (KSEG * CP1) / 4; i += BLK) {
        *reinterpret_cast<float4*>(&lm[i * 4]) =
            *reinterpret_cast<const float4*>(src + (size_t)i * 4);
    }
#endif
    __syncthreads();

    const long base = (long)bt * NPT + (long)blockIdx.x * (TILES * BLK);
    for (int t = 0; t < TILES; ++t) {
        const long p  = base + t * BLK + tid;
        const int  id = gid[p];
        float c[3];
        const bool sel = point_sel(id, bt, p, center, mean1, cnt1, c);

        float x[CH];
        if (sel) {
            const float* m = &lm[id * CP1];
#pragma unroll
            for (int q = 0; q < CH; ++q) x[q] = m[q];
        } else {
            if (id < 0 || id >= KSEG) {   // point_sel skipped the center load
                c[0] = center[p * 3 + 0];
                c[1] = center[p * 3 + 1];
                c[2] = center[p * 3 + 2];
            }
            x[0] = c[0]; x[1] = c[1]; x[2] = c[2];
            x[3] = offset[p * 3 + 0]; x[4] = offset[p * 3 + 1]; x[5] = offset[p * 3 + 2];
            x[6] = opacity[p];
            x[7] = scale[p * 3 + 0]; x[8] = scale[p * 3 + 1]; x[9] = scale[p * 3 + 2];
            const float4 r = *reinterpret_cast<const float4*>(rotation + p * 4);
            x[10] = r.x; x[11] = r.y; x[12] = r.z; x[13] = r.w;
            x[14] = feat_dc[p * 3 + 0]; x[15] = feat_dc[p * 3 + 1]; x[16] = feat_dc[p * 3 + 2];
            x[17] = confidence[p];
            const float4* a4 = reinterpret_cast<const float4*>(aff + p * 16);
            const float4* m4 = reinterpret_cast<const float4*>(motion + p * 16);
#pragma unroll
            for (int q = 0; q < 4; ++q) {
                const float4 av = a4[q];
                x[18 + q * 4 + 0] = av.x; x[18 + q * 4 + 1] = av.y;
                x[18 + q * 4 + 2] = av.z; x[18 + q * 4 + 3] = av.w;
            }
#pragma unroll
            for (int q = 0; q < 4; ++q) {
                const float4 mv = m4[q];
                x[34 + q * 4 + 0] = mv.x; x[34 + q * 4 + 1] = mv.y;
                x[34 + q * 4 + 2] = mv.z; x[34 + q * 4 + 3] = mv.w;
            }
        }
        float2* o2 = reinterpret_cast<float2*>(out + (size_t)p * CH);  // 8B aligned
#pragma unroll
        for (int q = 0; q < CH / 2; ++q) o2[q] = make_float2(x[2 * q], x[2 * q + 1]);
    }
}

extern "C" void kernel_launch(void* const* d_in, const int* in_sizes, int n_in,
                              void* d_out, int out_size, void* d_ws, size_t ws_size,
                              hipStream_t stream) {
    const float* center     = (const float*)d_in[0];
    const float* offset     = (const float*)d_in[1];
    const float* opacity    = (const float*)d_in[2];
    const float* scale      = (const float*)d_in[3];
    const float* rotation   = (const float*)d_in[4];
    const float* feat_dc    = (const float*)d_in[5];
    const float* confidence = (const float*)d_in[6];
    const float* aff        = (const float*)d_in[7];
    const float* motion     = (const float*)d_in[8];
    const int*   gid        = (const int*)d_in[9];
    float* ws  = (float*)d_ws;
    float* out = (float*)d_out;

    (void)in_sizes; (void)n_in; (void)out_size; (void)ws_size;

    k_zero<<<WS_ZERO / BLK, BLK, 0, stream>>>(ws);
    k_pass1<<<dim3(BLKX, BTN), BLK, 0, stream>>>(center, gid, ws + WS_CNT1, ws + WS_SUM1);
    k_mean1<<<(SEGS + BLK - 1) / BLK, BLK, 0, stream>>>(ws + WS_CNT1, ws + WS_SUM1, ws + WS_MEAN1);
    k_pass2<<<dim3(BLKX, BTN), BLK, 0, stream>>>(center, offset, opacity, scale, rotation,
                                                 feat_dc, confidence, aff, motion, gid,
                                                 ws + WS_MEAN1, ws + WS_CNT1, ws + WS_SUM2);
    k_mean2<<<SEGS, 64, 0, stream>>>(ws + WS_SUM2, ws + WS_MEAN2);
    k_out<<<dim3(BLKX, BTN), BLK, 0, stream>>>(center, offset, opacity, scale, rotation,
                                               feat_dc, confidence, aff, motion, gid,
                                               ws + WS_MEAN1, ws + WS_CNT1, ws + WS_MEAN2, out);
}